// SimpleNet_77240691851596
// MI455X (gfx1250) — compile-verified
//
#include <hip/hip_runtime.h>

// ---------------------------------------------------------------------------
// Fused Lux policy network for MI455X (gfx1250, wave32).
//
// Memory-bound (~230MB inputs, ~1 GFLOP): fuse the whole per-image pipeline
// in one workgroup using the 320KB/WGP LDS. The 12x16 head GEMM runs on
// V_WMMA_F32_16X16X4_F32 (exact f32, K=16 in 4 steps) over 16-pixel tiles.
// A-fragments are built with lane-half branches (no dynamic register
// indexing -> no v_cndmask select chains). Bool masks are read as u64
// (8 pixels/load, coalesced). Segment sums via global_atomic_add_f32.
// ---------------------------------------------------------------------------

typedef __attribute__((ext_vector_type(2))) float v2f;
typedef __attribute__((ext_vector_type(8))) float v8f;

#define NB   256
#define HW   2304   // 48*48
#define NG   1000
#define NEGV (-1000000000.0f)

// LDS partition (float offsets), total 51108 floats = 204432 bytes (dynamic)
#define OFF_M     0        // m        2*HW
#define OFF_P1    4608     // pool3(m) 2*HW
#define OFF_C5    9216     // conv5    8*HW
#define OFF_LD    27648    // pool3    8*HW
#define OFF_WLD   46080    // Wld      400
#define OFF_BALL  46480    // B matrix 16x16 (k-major)
#define OFF_BIASH 46736    // head bias 16
#define OFF_G     46752    // g0,g1 (+pad)
#define OFF_MASK  46756    // per-pixel mask words, 2304 u32
#define OFF_WT    49060    // per-wave D exchange tiles, 8*256
#define SMEM_FLOATS 51108

__device__ __forceinline__ float lrelu(float x) { return x > 0.0f ? x : 0.01f * x; }

// set bit `bit` in bb[j] for each of 8 bytes of acc that is non-zero
__device__ __forceinline__ void setbits64(unsigned long long acc, int bit, unsigned *bb) {
#pragma unroll
  for (int j = 0; j < 8; ++j)
    if ((acc >> (8 * j)) & 0xFFull) bb[j] |= (1u << bit);
}

// masked log-softmax stats: logp at argmax (first max), entropy
__device__ __forceinline__ void masked_cat(const float *lg, unsigned mask, int nch,
                                           float &logp, float &ent) {
  float ml[7];
  for (int a = 0; a < nch; ++a) ml[a] = ((mask >> a) & 1u) ? lg[a] : NEGV;
  float mx = ml[0]; int am = 0;
  for (int a = 1; a < nch; ++a) if (ml[a] > mx) { mx = ml[a]; am = a; }
  float se = 0.0f;
  for (int a = 0; a < nch; ++a) se += expf(ml[a] - mx);
  float lse = mx + logf(se);
  logp = ml[am] - lse;
  ent = 0.0f;
  for (int a = 0; a < nch; ++a) { float lp = ml[a] - lse; ent -= expf(lp) * lp; }
}

__global__ void zero_kernel(float *out, int n) {
  int i = blockIdx.x * blockDim.x + threadIdx.x;
  if (i < n) out[i] = 0.0f;
}

__global__ void __launch_bounds__(256)
policy_kernel(const float *gf, const float *mapf, const float *facf, const float *unitf,
              const int *locf,
              const unsigned char *va_fa, const unsigned char *va_mv,
              const unsigned char *va_tr, const unsigned char *va_pk,
              const unsigned char *va_dg, const unsigned char *va_sd,
              const unsigned char *va_rc, const unsigned char *va_dn,
              const float *Wg, const float *bg, const float *Wf, const float *bf,
              const float *Wu, const float *bu, const float *Wm, const float *bm,
              const float *Wld, const float *bld, const float *Wc, const float *bc,
              const float *Wfh, const float *bfh, const float *Wact, const float *bact,
              float *out) {
  extern __shared__ float sm[];
  const int b = blockIdx.x;
  const int tid = threadIdx.x;

  // ---- stage 0: small weights into LDS, g scalars -------------------------
  if (tid < 400) sm[OFF_WLD + tid] = Wld[tid];
  {
    int k = tid >> 4, n = tid & 15;     // B[k][n] = headW[n][k], 12 real heads
    float v = 0.0f;
    if (n < 4)        v = Wfh[n * 16 + k];
    else if (n < 11)  v = Wact[(n - 4) * 16 + k];
    else if (n == 11) v = Wc[k];
    sm[OFF_BALL + tid] = v;
  }
  if (tid < 16) {
    float v = 0.0f;
    if (tid < 4)        v = bfh[tid];
    else if (tid < 11)  v = bact[tid - 4];
    else if (tid == 11) v = bc[0];
    sm[OFF_BIASH + tid] = v;
  }
  if (tid < 2) {
    float v = gf[b * 2 + 0] * Wg[tid * 2 + 0] + gf[b * 2 + 1] * Wg[tid * 2 + 1] + bg[tid];
    sm[OFF_G + tid] = lrelu(v);
  }

  // ---- stage 1a: m = lrelu(conv1x1(map)) ----------------------------------
  {
    float wm[12], bm0 = bm[0], bm1 = bm[1];
#pragma unroll
    for (int i = 0; i < 12; ++i) wm[i] = Wm[i];
    for (int p = tid; p < HW; p += 256) {
      float x[6];
#pragma unroll
      for (int c = 0; c < 6; ++c) x[c] = mapf[(b * 6 + c) * HW + p];
      float m0 = bm0, m1 = bm1;
#pragma unroll
      for (int c = 0; c < 6; ++c) { m0 += x[c] * wm[c]; m1 += x[c] * wm[6 + c]; }
      sm[OFF_M + p]      = lrelu(m0);
      sm[OFF_M + HW + p] = lrelu(m1);
    }
  }

  // ---- stage 1b: mask bits (the big bandwidth pass), 8 pixels per u64 -----
  {
    typedef unsigned long long u64;
    const u64 *fa64 = (const u64 *)va_fa;
    const u64 *mv64 = (const u64 *)va_mv;
    const u64 *tr64 = (const u64 *)va_tr;
    const u64 *pk64 = (const u64 *)va_pk;
    const u64 *dg64 = (const u64 *)va_dg;
    const u64 *sd64 = (const u64 *)va_sd;
    const u64 *rc64 = (const u64 *)va_rc;
    const u64 *dn64 = (const u64 *)va_dn;
    unsigned *msk = (unsigned *)&sm[OFF_MASK];
    for (int gi = tid; gi < 288; gi += 256) {   // 288 groups of 8 pixels
      unsigned bb[8] = {0, 0, 0, 0, 0, 0, 0, 0};
      u64 acc;
      acc = 0;                                   // bit0: move (10 ch)
      for (int c = 0; c < 10; ++c) acc |= mv64[(b * 10 + c) * 288 + gi];
      setbits64(acc, 0, bb);
      acc = 0;                                   // bit1: transfer (250 ch)
#pragma unroll 5
      for (int c = 0; c < 250; ++c) acc |= tr64[(b * 250 + c) * 288 + gi];
      setbits64(acc, 1, bb);
      acc = 0;                                   // bit2: pickup (50 ch)
#pragma unroll 5
      for (int c = 0; c < 50; ++c) acc |= pk64[(b * 50 + c) * 288 + gi];
      setbits64(acc, 2, bb);
      acc = dg64[(b * 2 + 0) * 288 + gi] | dg64[(b * 2 + 1) * 288 + gi];  // bit3: dig
      setbits64(acc, 3, bb);
      acc = sd64[b * 288 + gi];                  // bit4: self destruct
      setbits64(acc, 4, bb);
      acc = rc64[(b * 2 + 0) * 288 + gi] | rc64[(b * 2 + 1) * 288 + gi];  // bit5
      setbits64(acc, 5, bb);
      acc = dn64[b * 288 + gi];                  // bit6: do nothing
      setbits64(acc, 6, bb);
#pragma unroll
      for (int c = 0; c < 4; ++c) {              // bits 8..11: per-ch factory
        acc = fa64[(b * 4 + c) * 288 + gi];
        setbits64(acc, 8 + c, bb);
      }
#pragma unroll
      for (int j = 0; j < 8; ++j) msk[gi * 8 + j] = bb[j];
    }
  }
  __syncthreads();

  // ---- stage 2: p1 = avg_pool3(m), zero-padded ----------------------------
  for (int p = tid; p < HW; p += 256) {
    int y = p / 48, x = p - y * 48;
    float s0 = 0.0f, s1 = 0.0f;
#pragma unroll
    for (int dy = -1; dy <= 1; ++dy) {
      int yy = y + dy; if ((unsigned)yy >= 48u) continue;
#pragma unroll
      for (int dx = -1; dx <= 1; ++dx) {
        int xx = x + dx; if ((unsigned)xx >= 48u) continue;
        int q = yy * 48 + xx;
        s0 += sm[OFF_M + q]; s1 += sm[OFF_M + HW + q];
      }
    }
    sm[OFF_P1 + p]      = s0 * (1.0f / 9.0f);
    sm[OFF_P1 + HW + p] = s1 * (1.0f / 9.0f);
  }
  __syncthreads();

  // ---- stage 3: c5 = lrelu(conv5x5(p1)), 2->8 ch, zero-padded -------------
  for (int p = tid; p < HW; p += 256) {
    int y = p / 48, x = p - y * 48;
    float acc[8];
#pragma unroll
    for (int o = 0; o < 8; ++o) acc[o] = bld[o];
#pragma unroll
    for (int i = 0; i < 2; ++i)
      for (int dy = 0; dy < 5; ++dy) {
        int yy = y + dy - 2; if ((unsigned)yy >= 48u) continue;
        for (int dx = 0; dx < 5; ++dx) {
          int xx = x + dx - 2; if ((unsigned)xx >= 48u) continue;
          float v = sm[OFF_P1 + i * HW + yy * 48 + xx];
#pragma unroll
          for (int o = 0; o < 8; ++o)
            acc[o] += v * sm[OFF_WLD + ((o * 2 + i) * 5 + dy) * 5 + dx];
        }
      }
#pragma unroll
    for (int o = 0; o < 8; ++o) sm[OFF_C5 + o * HW + p] = lrelu(acc[o]);
  }
  __syncthreads();

  // ---- stage 4: ld = avg_pool3(c5) ----------------------------------------
  for (int p = tid; p < HW; p += 256) {
    int y = p / 48, x = p - y * 48;
    float s[8];
#pragma unroll
    for (int o = 0; o < 8; ++o) s[o] = 0.0f;
#pragma unroll
    for (int dy = -1; dy <= 1; ++dy) {
      int yy = y + dy; if ((unsigned)yy >= 48u) continue;
#pragma unroll
      for (int dx = -1; dx <= 1; ++dx) {
        int xx = x + dx; if ((unsigned)xx >= 48u) continue;
        int q = yy * 48 + xx;
#pragma unroll
        for (int o = 0; o < 8; ++o) s[o] += sm[OFF_C5 + o * HW + q];
      }
    }
#pragma unroll
    for (int o = 0; o < 8; ++o) sm[OFF_LD + o * HW + p] = s[o] * (1.0f / 9.0f);
  }
  __syncthreads();

  // ---- stage 5: heads via WMMA f32 16x16x4 over 16-pixel tiles ------------
  const int wv = tid >> 5, ln = tid & 31;
  const int n = ln & 15;            // head column this lane carries in B/D
  const bool hi = (ln >= 16);       // lane half -> K offset (0 or 2) in A/B
  const int khalf = hi ? 2 : 0;

  float wfv[12], wuv[8];
#pragma unroll
  for (int i = 0; i < 12; ++i) wfv[i] = Wf[i];
#pragma unroll
  for (int i = 0; i < 8; ++i) wuv[i] = Wu[i];
  const float bf0 = bf[0], bf1 = bf[1], bu0 = bu[0], bu1 = bu[1];
  const float g0 = sm[OFF_G + 0], g1 = sm[OFF_G + 1];
  const float biasn = sm[OFF_BIASH + n];

  v2f bfrag[4];
#pragma unroll
  for (int s = 0; s < 4; ++s) {     // B 4x16 slice: VGPR0=row k0+khalf, VGPR1=+1
    int k = 4 * s + khalf;
    bfrag[s].x = sm[OFF_BALL + k * 16 + n];
    bfrag[s].y = sm[OFF_BALL + (k + 1) * 16 + n];
  }

  float *wt = &sm[OFF_WT + wv * 256];
  const unsigned *msk = (const unsigned *)&sm[OFF_MASK];

  for (int t = wv; t < 144; t += 8) {           // 144 tiles, 18 per wave
    const int pbase = t * 16;
    const int p = pbase + (ln & 15);

    // this lane's 8 A-matrix elements (K = 4s+khalf, 4s+khalf+1), constant
    // indices only -> pure registers, no select chains. Lane halves need
    // disjoint comb channels, so each value is produced exactly once/pixel.
    float e[8];
    if (hi) {
      // K = 2,3 (f), 6,7 (m), 10,11 (ld2,ld3), 14,15 (ld6,ld7)
      float a[6];
#pragma unroll
      for (int c = 0; c < 6; ++c) a[c] = facf[(b * 6 + c) * HW + p];
      float f0 = bf0, f1 = bf1;
#pragma unroll
      for (int c = 0; c < 6; ++c) { f0 += a[c] * wfv[c]; f1 += a[c] * wfv[6 + c]; }
      e[0] = lrelu(f0); e[1] = lrelu(f1);
      e[2] = sm[OFF_M + p]; e[3] = sm[OFF_M + HW + p];
      e[4] = sm[OFF_LD + 2 * HW + p]; e[5] = sm[OFF_LD + 3 * HW + p];
      e[6] = sm[OFF_LD + 6 * HW + p]; e[7] = sm[OFF_LD + 7 * HW + p];
    } else {
      // K = 0,1 (g), 4,5 (u), 8,9 (ld0,ld1), 12,13 (ld4,ld5)
      e[0] = g0; e[1] = g1;
      float a[4];
#pragma unroll
      for (int c = 0; c < 4; ++c) a[c] = unitf[(b * 4 + c) * HW + p];
      float u0 = bu0, u1 = bu1;
#pragma unroll
      for (int c = 0; c < 4; ++c) { u0 += a[c] * wuv[c]; u1 += a[c] * wuv[4 + c]; }
      e[2] = lrelu(u0); e[3] = lrelu(u1);
      e[4] = sm[OFF_LD + 0 * HW + p]; e[5] = sm[OFF_LD + 1 * HW + p];
      e[6] = sm[OFF_LD + 4 * HW + p]; e[7] = sm[OFF_LD + 5 * HW + p];
    }

    // C preloaded with head biases; 4 chained exact-f32 WMMAs (K=16 total)
    v8f acc;
#pragma unroll
    for (int v = 0; v < 8; ++v) acc[v] = biasn;
#pragma unroll
    for (int s = 0; s < 4; ++s) {
      v2f a; a.x = e[2 * s]; a.y = e[2 * s + 1];
      acc = __builtin_amdgcn_wmma_f32_16x16x4_f32(
          false, a, false, bfrag[s], (short)0, acc, false, false);
    }

    // D layout: VGPR v holds rows v (lanes 0-15) / v+8 (lanes 16-31), col n
    const int rbase = hi ? 8 : 0;
#pragma unroll
    for (int v = 0; v < 8; ++v) wt[(rbase + v) * 16 + n] = acc[v];
    asm volatile("s_wait_dscnt 0" ::: "memory");   // same-wave LDS RAW fence
    __builtin_amdgcn_wave_barrier();

    // epilogue: lanes 0-15 -> factory head, lanes 16-31 -> unit head,
    // both for pixel pbase + (ln&15)
    {
      const int r = ln & 15;
      const unsigned mk = msk[pbase + r];
      float *ob = out + b * 3 * NG;
      if (!hi) {
        const unsigned fm = (mk >> 8) & 0xFu;
        if (fm) {
          float hv[4];
#pragma unroll
          for (int i = 0; i < 4; ++i) hv[i] = wt[r * 16 + i];
          const float critic = wt[r * 16 + 11];
          const int fid = locf[(b * 2 + 0) * HW + pbase + r];
          float logp, ent;
          masked_cat(hv, fm, 4, logp, ent);
          atomicAdd(&ob[fid], logp);
          atomicAdd(&ob[NG + fid], critic);
          atomicAdd(&ob[2 * NG + fid], ent);
        }
      } else {
        const unsigned um = mk & 0x7Fu;
        if (um) {
          float hv[7];
#pragma unroll
          for (int i = 0; i < 7; ++i) hv[i] = wt[r * 16 + 4 + i];
          const float critic = wt[r * 16 + 11];
          const int uid = locf[(b * 2 + 1) * HW + pbase + r];
          float logp, ent;
          masked_cat(hv, um, 7, logp, ent);
          atomicAdd(&ob[uid], logp);
          atomicAdd(&ob[NG + uid], critic);
          atomicAdd(&ob[2 * NG + uid], ent);
        }
      }
    }
    __builtin_amdgcn_wave_barrier();
  }
}

extern "C" void kernel_launch(void *const *d_in, const int *in_sizes, int n_in,
                              void *d_out, int out_size, void *d_ws, size_t ws_size,
                              hipStream_t stream) {
  (void)in_sizes; (void)n_in; (void)d_ws; (void)ws_size; (void)out_size;
  float *out = (float *)d_out;
  const int nv = NB * 3 * NG;                 // 768000
  zero_kernel<<<(nv + 255) / 256, 256, 0, stream>>>(out, nv);

  policy_kernel<<<NB, 256, SMEM_FLOATS * sizeof(float), stream>>>(
      (const float *)d_in[0],          // global_feature
      (const float *)d_in[1],          // map_feature
      (const float *)d_in[2],          // factory_feature
      (const float *)d_in[3],          // unit_feature
      (const int *)d_in[4],            // location_feature
      (const unsigned char *)d_in[5],  // va_factory_act (bool)
      (const unsigned char *)d_in[6],  // va_move
      (const unsigned char *)d_in[7],  // va_transfer
      (const unsigned char *)d_in[8],  // va_pickup
      (const unsigned char *)d_in[9],  // va_dig
      (const unsigned char *)d_in[10], // va_self_destruct
      (const unsigned char *)d_in[11], // va_recharge
      (const unsigned char *)d_in[12], // va_do_nothing
      (const float *)d_in[13], (const float *)d_in[14],   // Wg, bg
      (const float *)d_in[15], (const float *)d_in[16],   // Wf, bf
      (const float *)d_in[17], (const float *)d_in[18],   // Wu, bu
      (const float *)d_in[19], (const float *)d_in[20],   // Wm, bm
      (const float *)d_in[21], (const float *)d_in[22],   // Wld, bld
      (const float *)d_in[23], (const float *)d_in[24],   // Wc, bc
      (const float *)d_in[25], (const float *)d_in[26],   // Wfh, bfh
      (const float *)d_in[27], (const float *)d_in[28],   // Wact, bact
      out);
}